// GConv_20418274525425
// MI455X (gfx1250) — compile-verified
//
#include <hip/hip_runtime.h>

#define N_NODES 50000
#define D       128
#define N_EDGES 800000

typedef float v2f __attribute__((ext_vector_type(2)));
typedef float v8f __attribute__((ext_vector_type(8)));

// ---------------------------------------------------------------------------
// Kernel A: zero the workspace (agg[N*D] + deg[N]) — must run every launch.
// ---------------------------------------------------------------------------
__global__ void zero_ws_kernel(float* __restrict__ p, int n) {
    int i = blockIdx.x * blockDim.x + threadIdx.x;
    int stride = gridDim.x * blockDim.x;
    for (; i < n; i += stride) p[i] = 0.0f;
}

// ---------------------------------------------------------------------------
// Kernel B: scatter-add  agg[dst,:] += x[src,:]  and  deg[dst] += 1.
// One thread per (edge, feature). x is L2-resident (25.6MB << 192MB L2), so
// the random gather hits L2; atomics are serviced by the L2 atomic units.
// ---------------------------------------------------------------------------
__global__ void scatter_add_kernel(const float* __restrict__ x,
                                   const long long* __restrict__ ei,
                                   float* __restrict__ agg,
                                   float* __restrict__ deg) {
    long long total  = (long long)N_EDGES * D;
    long long stride = (long long)gridDim.x * blockDim.x;
    for (long long t = (long long)blockIdx.x * blockDim.x + threadIdx.x;
         t < total; t += stride) {
        int e = (int)(t >> 7);        // edge id
        int f = (int)(t & (D - 1));   // feature id
        int src = (int)ei[e];
        int dst = (int)ei[N_EDGES + e];
        atomicAdd(&agg[(long long)dst * D + f], x[(long long)src * D + f]);
        if (f == 0) atomicAdd(&deg[dst], 1.0f);
    }
}

// ---------------------------------------------------------------------------
// Kernel C: out = relu( (agg/deg) @ Wl + x @ Wr + b )  via V_WMMA_F32_16X16X4_F32
//
// Block = 256 threads = 8 waves. Wave w owns rows [blk*128 + w*16, +16) and all
// 128 output columns: 8 accumulators of v8f (16x16 f32 C/D tiles).
// Wl then Wr staged through one 64KB LDS buffer (barrier between phases),
// stored K-PAIR-INTERLEAVED:  sW2[(k/2)*128 + n] = { W[k][n], W[k+1][n] }
// so each B fragment is a single ds_load_b64 straight into a VGPR pair
// (no shuffle movs before the WMMA).
//
// f32 A-fragment (16x4, MxK): lanes 0-15 hold M=lane, VGPR0=K0,VGPR1=K1;
//                             lanes 16-31 hold M=lane-16, VGPR0=K2,VGPR1=K3.
// f32 B-fragment (4x16, KxN): lanes hold N; VGPR/lane-half selects K (mirror).
// C/D tile: VGPR r, lanes 0-15 -> M=r, lanes 16-31 -> M=8+r; N = lane%16.
// ---------------------------------------------------------------------------
__global__ __launch_bounds__(256)
void sage_gemm_kernel(const float* __restrict__ agg,
                      const float* __restrict__ deg,
                      const float* __restrict__ x,
                      const float* __restrict__ Wl,
                      const float* __restrict__ Wr,
                      const float* __restrict__ bias,
                      float* __restrict__ out) {
    // 64 KB: (D/2) k-pairs x D columns of float2
    __shared__ __align__(16) v2f sW2[(D / 2) * D];

    const int tid   = threadIdx.x;
    const int wave  = tid >> 5;
    const int lane  = tid & 31;
    const int l16   = lane & 15;
    const int kh    = lane >> 4;               // which K pair this half-wave holds
    const int mbase = blockIdx.x * 128 + wave * 16;

    // Row this lane supplies for A fragments (A layout: lanes = M).
    const int  arow   = mbase + l16;
    const bool rowok  = arow < N_NODES;
    const int  arow_c = rowok ? arow : (N_NODES - 1);   // clamped, branch-free loads

    // Accumulators initialized with bias (bias depends only on column = lane).
    v8f acc[8];
#pragma unroll
    for (int n = 0; n < 8; ++n) {
        float bv = bias[n * 16 + l16];
        acc[n] = (v8f){bv, bv, bv, bv, bv, bv, bv, bv};
    }

    // 1/max(deg,1), folded into phase-1 A loads; zero-mask for out-of-range rows.
    const float dval   = deg[arow_c];
    const float invdeg = rowok ? (1.0f / fmaxf(dval, 1.0f)) : 0.0f;
    const float xmask  = rowok ? 1.0f : 0.0f;

    // Per-lane base index into sW2 (float2 units): kh selects which k-pair of
    // the 4-wide K step this half-wave consumes.
    const int bbase = kh * D + l16;

    // ---------------- Phase 1: acc += (agg * invdeg) @ Wl ----------------
    // Stage Wl K-pair-interleaved. Coalesced global reads (consecutive tid ->
    // consecutive columns of one row), contiguous ds_store_b64 writes.
    for (int i = tid; i < (D / 2) * D; i += 256) {
        int kp = i >> 7;          // k-pair index
        int n  = i & (D - 1);     // column
        v2f w;
        w.x = Wl[(2 * kp) * D + n];
        w.y = Wl[(2 * kp + 1) * D + n];
        sW2[i] = w;
    }
    __syncthreads();

    {
        const float* Arow = agg + (long long)arow_c * D;
#pragma unroll 4
        for (int k0 = 0; k0 < D; k0 += 4) {
            const int ka = k0 + 2 * kh;
            v2f a;
            a.x = Arow[ka]     * invdeg;
            a.y = Arow[ka + 1] * invdeg;
            const int bk = (k0 >> 1) * D + bbase;
#pragma unroll
            for (int n = 0; n < 8; ++n) {
                v2f bf = sW2[bk + n * 16];          // one ds_load_b64
                acc[n] = __builtin_amdgcn_wmma_f32_16x16x4_f32(
                    false, a, false, bf, (short)0, acc[n], false, false);
            }
        }
    }
    __syncthreads();

    // ---------------- Phase 2: acc += x @ Wr ----------------
    for (int i = tid; i < (D / 2) * D; i += 256) {
        int kp = i >> 7;
        int n  = i & (D - 1);
        v2f w;
        w.x = Wr[(2 * kp) * D + n];
        w.y = Wr[(2 * kp + 1) * D + n];
        sW2[i] = w;
    }
    __syncthreads();

    {
        const float* Xrow = x + (long long)arow_c * D;
#pragma unroll 4
        for (int k0 = 0; k0 < D; k0 += 4) {
            const int ka = k0 + 2 * kh;
            v2f a;
            a.x = Xrow[ka]     * xmask;
            a.y = Xrow[ka + 1] * xmask;
            const int bk = (k0 >> 1) * D + bbase;
#pragma unroll
            for (int n = 0; n < 8; ++n) {
                v2f bf = sW2[bk + n * 16];          // one ds_load_b64
                acc[n] = __builtin_amdgcn_wmma_f32_16x16x4_f32(
                    false, a, false, bf, (short)0, acc[n], false, false);
            }
        }
    }

    // ---------------- ReLU + store (C/D layout: row = mbase + 8*kh + r) ----
    const int rbase = mbase + 8 * kh;
#pragma unroll
    for (int n = 0; n < 8; ++n) {
        const int col = n * 16 + l16;
#pragma unroll
        for (int r = 0; r < 8; ++r) {
            const int orow = rbase + r;
            if (orow < N_NODES) {
                float v = acc[n][r];
                out[(long long)orow * D + col] = v > 0.0f ? v : 0.0f;
            }
        }
    }
}

// ---------------------------------------------------------------------------
// kernel_launch
//   d_in[0] = x        [N*D]   f32
//   d_in[1] = edge_idx [2*E]   i64
//   d_in[2] = W_l      [D*D]   f32
//   d_in[3] = W_r      [D*D]   f32
//   d_in[4] = b        [D]     f32
//   d_out   = out      [N*D]   f32
//   d_ws    : agg[N*D] f32, deg[N] f32
// ---------------------------------------------------------------------------
extern "C" void kernel_launch(void* const* d_in, const int* in_sizes, int n_in,
                              void* d_out, int out_size, void* d_ws, size_t ws_size,
                              hipStream_t stream) {
    const float*     x   = (const float*)d_in[0];
    const long long* ei  = (const long long*)d_in[1];
    const float*     Wl  = (const float*)d_in[2];
    const float*     Wr  = (const float*)d_in[3];
    const float*     b   = (const float*)d_in[4];
    float*           out = (float*)d_out;

    float* agg = (float*)d_ws;              // N*D floats
    float* deg = agg + (size_t)N_NODES * D; // N floats

    // 1) zero workspace (every call — deterministic under graph replay)
    {
        int n = N_NODES * (D + 1);
        int blocks = (n + 255) / 256;
        zero_ws_kernel<<<blocks, 256, 0, stream>>>(agg, n);
    }

    // 2) scatter-add aggregation + degrees
    {
        long long total = (long long)N_EDGES * D;   // 102.4M
        int blocks = (int)((total + 255) / 256);    // 400000 blocks
        scatter_add_kernel<<<blocks, 256, 0, stream>>>(x, ei, agg, deg);
    }

    // 3) fused mean + dual GEMM + bias + ReLU (f32 WMMA)
    {
        int blocks = (N_NODES + 127) / 128;         // 391
        sage_gemm_kernel<<<blocks, 256, 0, stream>>>(agg, deg, x, Wl, Wr, b, out);
    }
}